// BallLoss_9045201126030
// MI455X (gfx1250) — compile-verified
//
#include <hip/hip_runtime.h>

// CDNA5 / gfx1250, wave32. Fp32 GEMM via V_WMMA_F32_16X16X4_F32.

typedef __attribute__((ext_vector_type(2))) float v2f;
typedef __attribute__((ext_vector_type(4))) float v4f;
typedef __attribute__((ext_vector_type(8))) float v8f;

#define N_ROWS 65536
#define N_CTR  2048
#define DIM    64
#define EPSF   1e-12f

#define WAVES_PER_BLOCK 8
#define BLOCK_THREADS   256          // 8 wave32
#define ROWS_PER_BLOCK  (WAVES_PER_BLOCK * 16)      // 128
#define NUM_BLOCKS      (N_ROWS / ROWS_PER_BLOCK)   // 512
#define COL_TILES       (N_CTR / 16)                // 128
#define LDS_STRIDE      68           // 64 + 4 pad -> conflict-free ds_load_b64
#define TILE_FLOATS     (16 * LDS_STRIDE)           // one buffered tile

// ---------------------------------------------------------------- |c_j|^2
__global__ void cn2_kernel(const float* __restrict__ C, float* __restrict__ cn2)
{
    int j = blockIdx.x * blockDim.x + threadIdx.x;
    if (j < N_CTR) {
        const v4f* p = (const v4f*)(C + (size_t)j * DIM);
        float s = 0.f;
        #pragma unroll
        for (int q = 0; q < DIM / 4; ++q) {
            v4f v = p[q];
            s += v.x * v.x + v.y * v.y + v.z * v.z + v.w * v.w;
        }
        cn2[j] = s;
    }
}

// ---------------------------------------------------------------- main GEMM+loss
__global__ __launch_bounds__(BLOCK_THREADS)
void ball_main(const float* __restrict__ E,
               const float* __restrict__ C,
               const int*   __restrict__ labels,
               const float* __restrict__ cn2,
               double*      __restrict__ partials)
{
    __shared__ float  ldsB[2 * TILE_FLOATS];   // double-buffered center tile
    __shared__ double wsum[WAVES_PER_BLOCK];

    const int tid  = threadIdx.x;
    const int wave = tid >> 5;
    const int lane = tid & 31;
    const int l16  = lane & 15;
    const int h    = lane >> 4;          // selects K pair within group of 4

    const int rowbase = blockIdx.x * ROWS_PER_BLOCK + wave * 16;

    // cooperative-stage indices: 256 threads cover 16 rows x 16 float4 chunks
    const int cr = tid >> 4;             // center row within tile
    const int cs = tid & 15;             // float4 chunk
    const int lds_wr = cr * LDS_STRIDE + cs * 4;

    // --- A panel in registers: a[kk] = e[rowbase+l16][4kk+2h .. 4kk+2h+1]
    v2f a[16];
    const float* erow = E + (size_t)(rowbase + l16) * DIM;
    #pragma unroll
    for (int kk = 0; kk < 16; ++kk)
        a[kk] = *(const v2f*)(erow + 4 * kk + 2 * h);

    // --- |e|^2 of row l16 (lane halves hold K%4 in {2h, 2h+1})
    float p2 = 0.f;
    #pragma unroll
    for (int kk = 0; kk < 16; ++kk)
        p2 += a[kk].x * a[kk].x + a[kk].y * a[kk].y;
    float en2_full = p2 + __shfl_xor(p2, 16);

    // --- d_i: distance of row l16 to its own center (same quadratic expansion)
    int lab = labels[rowbase + l16];
    const float* crow = C + (size_t)lab * DIM;
    float pd = 0.f;
    #pragma unroll
    for (int kk = 0; kk < 16; ++kk) {
        v2f cv = *(const v2f*)(crow + 4 * kk + 2 * h);
        pd += a[kk].x * cv.x + a[kk].y * cv.y;
    }
    float dot_full = pd + __shfl_xor(pd, 16);
    float d2i      = en2_full + cn2[lab] - 2.f * dot_full;
    float di_full  = __builtin_amdgcn_sqrtf(fmaxf(d2i, EPSF));

    // --- redistribute to C/D output mapping: reg r <-> row 8h + r
    float enr[8], dir_[8];
    #pragma unroll
    for (int r = 0; r < 8; ++r) {
        enr[r]  = __shfl(en2_full, 8 * h + r);
        dir_[r] = __shfl(di_full,  8 * h + r);
    }

    // --- prologue: stage tile 0
    {
        v4f v = *(const v4f*)(C + (size_t)cr * DIM + cs * 4);
        *(v4f*)(&ldsB[lds_wr]) = v;
    }
    __syncthreads();

    double lacc = 0.0;

    for (int ct = 0; ct < COL_TILES; ++ct) {
        const bool  hasnext = (ct + 1) < COL_TILES;
        const float* curb   = &ldsB[(ct & 1) * TILE_FLOATS];

        // prefetch next tile from global early (latency hides behind WMMAs)
        v4f nv;
        if (hasnext)
            nv = *(const v4f*)(C + (size_t)((ct + 1) * 16 + cr) * DIM + cs * 4);

        float cn2v = cn2[ct * 16 + l16];

        v8f acc = {0.f, 0.f, 0.f, 0.f, 0.f, 0.f, 0.f, 0.f};
        #pragma unroll
        for (int kk = 0; kk < 16; ++kk) {
            v2f b = *(const v2f*)(curb + l16 * LDS_STRIDE + 4 * kk + 2 * h);
            acc = __builtin_amdgcn_wmma_f32_16x16x4_f32(
                      false, a[kk], false, b, (short)0, acc, false, false);
        }

        // epilogue: relu(d_i - dist), tile-local f32 sum, one f64 add per tile
        float tsum = 0.f;
        #pragma unroll
        for (int r = 0; r < 8; ++r) {
            float d2   = enr[r] + cn2v - 2.f * acc[r];
            float dist = __builtin_amdgcn_sqrtf(fmaxf(d2, EPSF));
            tsum += fmaxf(dir_[r] - dist, 0.f);
        }
        lacc += (double)tsum;

        // stage next tile into the other buffer, then single barrier
        if (hasnext)
            *(v4f*)(&ldsB[((ct + 1) & 1) * TILE_FLOATS + lds_wr]) = nv;
        __syncthreads();
    }

    // wave32 reduce, then block reduce, write per-block partial (deterministic)
    #pragma unroll
    for (int off = 16; off > 0; off >>= 1)
        lacc += __shfl_down(lacc, off);
    if (lane == 0) wsum[wave] = lacc;
    __syncthreads();
    if (tid == 0) {
        double s = 0.0;
        #pragma unroll
        for (int w = 0; w < WAVES_PER_BLOCK; ++w) s += wsum[w];
        partials[blockIdx.x] = s;
    }
}

// ---------------------------------------------------------------- final reduce
__global__ void finalize_kernel(const double* __restrict__ partials,
                                float* __restrict__ out)
{
    __shared__ double sh[256];
    int t = threadIdx.x;
    double s = 0.0;
    for (int i = t; i < NUM_BLOCKS; i += 256) s += partials[i];
    sh[t] = s;
    __syncthreads();
    for (int st = 128; st > 0; st >>= 1) {
        if (t < st) sh[t] += sh[t + st];
        __syncthreads();
    }
    if (t == 0) out[0] = (float)sh[0];
}

// ---------------------------------------------------------------- launcher
extern "C" void kernel_launch(void* const* d_in, const int* in_sizes, int n_in,
                              void* d_out, int out_size, void* d_ws, size_t ws_size,
                              hipStream_t stream)
{
    const float* E      = (const float*)d_in[0];   // [65536,64]
    const float* C      = (const float*)d_in[1];   // [2048,64]
    const int*   labels = (const int*)  d_in[2];   // [65536]
    float*       out    = (float*)d_out;

    double* partials = (double*)d_ws;                                  // 512 * 8B
    float*  cn2      = (float*)((char*)d_ws + NUM_BLOCKS * sizeof(double));

    cn2_kernel<<<N_CTR / 256, 256, 0, stream>>>(C, cn2);
    ball_main<<<NUM_BLOCKS, BLOCK_THREADS, 0, stream>>>(E, C, labels, cn2, partials);
    finalize_kernel<<<1, 256, 0, stream>>>(partials, out);
}